// Edge_GNNNet_58342835748897
// MI455X (gfx1250) — compile-verified
//
#include <hip/hip_runtime.h>
#include <math.h>

// ---------------- types ----------------
typedef float v2f __attribute__((ext_vector_type(2)));
typedef float v8f __attribute__((ext_vector_type(8)));

#define N_NODES 177
#define FEAT 128
#define OUT_C 128
#define NEG_SLOPE 0.2f
#define MTILES 12           // ceil(177/16)
#define S_PAD 180           // S row stride, padded to multiple of 4 (zero-filled)
#define H_ROWS 180          // h padded rows (rows 177..179 zeroed at init)

// workspace layout (float words)
#define OFF_H    0                                 // h: 180x128 (padded)
#define OFF_A    (OFF_H + H_ROWS * FEAT)           // a[dst] scalars
#define OFF_B    (OFF_A + N_NODES)                 // b[src] scalars
#define OFF_ESUM (OFF_B + N_NODES)
#define OFF_T    (OFF_ESUM + N_NODES)
#define OFF_AMAX (OFF_T + N_NODES)                 // ordered-uint float max
#define OFF_S    (OFF_AMAX + N_NODES)              // S: 177x180 scatter matrix (padded)
#define WS_FLOATS (OFF_S + N_NODES * S_PAD)        // ~55.8K floats (~224 KB)

// -------- ordered-uint float mapping for atomicMax on signed floats --------
__device__ __forceinline__ unsigned f2ord(float f) {
  unsigned u = __float_as_uint(f);
  return (u & 0x80000000u) ? ~u : (u | 0x80000000u);
}
__device__ __forceinline__ float ord2f(unsigned u) {
  return __uint_as_float((u & 0x80000000u) ? (u ^ 0x80000000u) : ~u);
}
#define ORD_NEG_INF 0x007FFFFFu   // f2ord(-inf)

// -------- fp32 WMMA fragment loaders (16x16x4, wave32), all unguarded --------
// A 16x4: lanes 0-15 hold rows M; VGPR0={K=k0 | K=k0+2}, VGPR1={K=k0+1 | K=k0+3}
// Row index is CLAMPED: garbage rows only feed store-masked output rows.
__device__ __forceinline__ v2f load_a_clamp(const float* __restrict__ A, int m0, int k0,
                                            int lane, int ld) {
  int m = m0 + (lane & 15);
  m = (m < N_NODES) ? m : (N_NODES - 1);
  int kb = k0 + ((lane >> 4) << 1);
  const float* p = A + m * ld + kb;
  v2f a; a.x = p[0]; a.y = p[1];
  return a;
}
// B 4x16: lanes 0-15 cols N; VGPR0={K=k0 | K=k0+2}, VGPR1={K=k0+1 | K=k0+3}
__device__ __forceinline__ v2f load_b(const float* __restrict__ B, int k0, int n0,
                                      int lane, int ld) {
  int n  = n0 + (lane & 15);
  int kb = k0 + ((lane >> 4) << 1);
  const float* p = B + kb * ld + n;
  v2f b; b.x = p[0]; b.y = p[ld];
  return b;
}
// A fragment from a 16x128 LDS tile (all rows valid)
__device__ __forceinline__ v2f load_a_lds(const float* A, int k0, int lane) {
  int m  = lane & 15;
  int kb = k0 + ((lane >> 4) << 1);
  v2f a;
  a.x = A[m * OUT_C + kb];
  a.y = A[m * OUT_C + kb + 1];
  return a;
}

// ---------------- K0: init scatter buffers + zero paddings ----------------
__global__ void k_init(float* __restrict__ ws) {
  int tid    = blockIdx.x * blockDim.x + threadIdx.x;
  int stride = gridDim.x * blockDim.x;
  unsigned* amax = (unsigned*)(ws + OFF_AMAX);
  for (int j = tid; j < N_NODES * S_PAD; j += stride) ws[OFF_S + j] = 0.0f;
  for (int j = tid; j < (H_ROWS - N_NODES) * FEAT; j += stride)      // h pad rows 177..179
    ws[OFF_H + N_NODES * FEAT + j] = 0.0f;
  for (int j = tid; j < N_NODES; j += stride) {
    ws[OFF_ESUM + j] = 0.0f;
    ws[OFF_T + j]    = 0.0f;
    amax[j]          = ORD_NEG_INF;
  }
}

// ---------------- K1: h = x @ W  (fp32 WMMA) ----------------
// grid = MTILES blocks of 128 threads (4 waves); wave w -> 32-col slab
__global__ void k_node_gemm(const float* __restrict__ x, const float* __restrict__ W,
                            float* __restrict__ ws) {
  float* h  = ws + OFF_H;
  int lane  = threadIdx.x & 31;
  int wave  = threadIdx.x >> 5;
  int m0    = blockIdx.x * 16;
  int n0    = wave * 32;
  v8f c0 = {}; v8f c1 = {};
  for (int k0 = 0; k0 < FEAT; k0 += 4) {
    v2f a  = load_a_clamp(x, m0, k0, lane, FEAT);
    v2f b0 = load_b(W, k0, n0,      lane, OUT_C);
    v2f b1 = load_b(W, k0, n0 + 16, lane, OUT_C);
    c0 = __builtin_amdgcn_wmma_f32_16x16x4_f32(false, a, false, b0, (short)0, c0, false, false);
    c1 = __builtin_amdgcn_wmma_f32_16x16x4_f32(false, a, false, b1, (short)0, c1, false, false);
  }
  int nb = n0 + (lane & 15);
  int mh = (lane >> 4) * 8;
  for (int j = 0; j < 8; ++j) {
    int m = m0 + j + mh;
    if (m < N_NODES) {
      h[m * OUT_C + nb]      = c0[j];
      h[m * OUT_C + nb + 16] = c1[j];
    }
  }
}

// ---------------- K2: per-node attention scalars a[n], b[n] ----------------
// one wave per node: a[n]=h[n]·att[0:128], b[n]=h[n]·att[128:256]
__global__ void k_ab(const float* __restrict__ att, float* __restrict__ ws) {
  const float* h = ws + OFF_H;
  int n = blockIdx.x, lane = threadIdx.x;
  float pa = 0.0f, pb = 0.0f;
  for (int i = lane; i < FEAT; i += 32) {
    float hv = h[n * OUT_C + i];
    pa += hv * att[i];
    pb += hv * att[OUT_C + i];
  }
  for (int m = 16; m >= 1; m >>= 1) {
    pa += __shfl_xor(pa, m, 32);
    pb += __shfl_xor(pb, m, 32);
  }
  if (lane == 0) { ws[OFF_A + n] = pa; ws[OFF_B + n] = pb; }
}

// ---------------- K3: segment max over dst ----------------
__global__ void k_edge_max(const int* __restrict__ ei, const float* __restrict__ ea,
                           const float* __restrict__ att, float* __restrict__ ws, int E) {
  const float* a = ws + OFF_A;
  const float* b = ws + OFF_B;
  unsigned* amax = (unsigned*)(ws + OFF_AMAX);
  float cc = att[2 * OUT_C];
  int tid = blockIdx.x * blockDim.x + threadIdx.x;
  int stride = gridDim.x * blockDim.x;
  for (int e = tid; e < E; e += stride) {
    int s = ei[e], d = ei[E + e];
    float al = a[d] + b[s] + cc * ea[e];
    al = (al > 0.0f) ? al : NEG_SLOPE * al;
    atomicMax(&amax[d], f2ord(al));
  }
}

// ---------------- K4: exp + scatter-accumulate S, esum, t ----------------
__global__ void k_edge_exp(const int* __restrict__ ei, const float* __restrict__ ea,
                           const float* __restrict__ att, float* __restrict__ ws, int E) {
  const float* a = ws + OFF_A;
  const float* b = ws + OFF_B;
  const unsigned* amax = (const unsigned*)(ws + OFF_AMAX);
  float* esum = ws + OFF_ESUM;
  float* t    = ws + OFF_T;
  float* S    = ws + OFF_S;
  float cc = att[2 * OUT_C];
  int tid = blockIdx.x * blockDim.x + threadIdx.x;
  int stride = gridDim.x * blockDim.x;
  for (int e = tid; e < E; e += stride) {
    int s = ei[e], d = ei[E + e];
    float eav = ea[e];
    float al = a[d] + b[s] + cc * eav;
    al = (al > 0.0f) ? al : NEG_SLOPE * al;
    float w = __expf(al - ord2f(amax[d]));
    atomicAdd(&esum[d], w);
    atomicAdd(&t[d], w * eav);
    atomicAdd(&S[d * S_PAD + s], w);
  }
}

// ---------------- K5: fused  out = (diag(1/esum)·(S@h)) @ U[:128] + t_n⊗U[128] + bias ----
// one block per 16-row strip; GEMM1 result normalized into LDS, GEMM2 reads it as A.
__global__ void k_final(const float* __restrict__ U, const float* __restrict__ bias,
                        const float* __restrict__ ws, float* __restrict__ out) {
  __shared__ float aggr[16 * OUT_C];   // 8 KB
  const float* h    = ws + OFF_H;      // padded to 180 rows (177..179 zero)
  const float* S    = ws + OFF_S;      // 177x180, padded cols zero
  const float* esum = ws + OFF_ESUM;
  const float* t    = ws + OFF_T;
  int lane = threadIdx.x & 31;
  int wave = threadIdx.x >> 5;
  int m0 = blockIdx.x * 16;
  int n0 = wave * 32;

  // GEMM1: aggr_strip = S[m0:m0+16, :180] @ h[:180, :128]   (K zero-padded to 180)
  v8f c0 = {}; v8f c1 = {};
  for (int k0 = 0; k0 < S_PAD; k0 += 4) {
    v2f a  = load_a_clamp(S, m0, k0, lane, S_PAD);
    v2f b0 = load_b(h, k0, n0,      lane, OUT_C);
    v2f b1 = load_b(h, k0, n0 + 16, lane, OUT_C);
    c0 = __builtin_amdgcn_wmma_f32_16x16x4_f32(false, a, false, b0, (short)0, c0, false, false);
    c1 = __builtin_amdgcn_wmma_f32_16x16x4_f32(false, a, false, b1, (short)0, c1, false, false);
  }
  // normalize rows by 1/(esum+eps), stage in LDS
  int nb = n0 + (lane & 15);
  int mh = (lane >> 4) * 8;
  for (int j = 0; j < 8; ++j) {
    int mloc = j + mh;
    int m = m0 + mloc;
    int mc = (m < N_NODES) ? m : (N_NODES - 1);
    float r = 1.0f / (esum[mc] + 1e-16f);
    aggr[mloc * OUT_C + nb]      = c0[j] * r;
    aggr[mloc * OUT_C + nb + 16] = c1[j] * r;
  }
  __syncthreads();

  // GEMM2: out_strip = aggr_strip @ U[0:128, 0:128]
  v8f d0 = {}; v8f d1 = {};
  for (int k0 = 0; k0 < OUT_C; k0 += 4) {
    v2f a  = load_a_lds(aggr, k0, lane);
    v2f b0 = load_b(U, k0, n0,      lane, OUT_C);
    v2f b1 = load_b(U, k0, n0 + 16, lane, OUT_C);
    d0 = __builtin_amdgcn_wmma_f32_16x16x4_f32(false, a, false, b0, (short)0, d0, false, false);
    d1 = __builtin_amdgcn_wmma_f32_16x16x4_f32(false, a, false, b1, (short)0, d1, false, false);
  }
  // epilogue: + t_norm[m] * U[128, n] + bias[n]
  for (int j = 0; j < 8; ++j) {
    int m = m0 + j + mh;
    if (m < N_NODES) {
      float tn = t[m] / (esum[m] + 1e-16f);
      out[m * OUT_C + nb]      = d0[j] + tn * U[OUT_C * OUT_C + nb]      + bias[nb];
      out[m * OUT_C + nb + 16] = d1[j] + tn * U[OUT_C * OUT_C + nb + 16] + bias[nb + 16];
    }
  }
}

// ---------------- host launch ----------------
extern "C" void kernel_launch(void* const* d_in, const int* in_sizes, int n_in,
                              void* d_out, int out_size, void* d_ws, size_t ws_size,
                              hipStream_t stream) {
  const float* x    = (const float*)d_in[0];   // [1,177,128]
  const float* ea   = (const float*)d_in[1];   // [E,1]
  const float* W    = (const float*)d_in[2];   // [128,128]
  const float* att  = (const float*)d_in[3];   // [1,1,257]
  const float* U    = (const float*)d_in[4];   // [129,128]
  const float* bias = (const float*)d_in[5];   // [128]
  const int*   ei   = (const int*)d_in[6];     // [2,E] int32
  int E = in_sizes[1];                         // edge_attr count == E (EDGE_DIM=1)
  float* ws  = (float*)d_ws;
  float* out = (float*)d_out;

  k_init<<<64, 256, 0, stream>>>(ws);
  k_node_gemm<<<MTILES, 128, 0, stream>>>(x, W, ws);
  k_ab<<<N_NODES, 32, 0, stream>>>(att, ws);
  k_edge_max<<<1024, 256, 0, stream>>>(ei, ea, att, ws, E);
  k_edge_exp<<<1024, 256, 0, stream>>>(ei, ea, att, ws, E);
  k_final<<<MTILES, 128, 0, stream>>>(U, bias, ws, out);
}